// LengthRegulator_5781025980416
// MI455X (gfx1250) — compile-verified
//
#include <hip/hip_runtime.h>
#include <stdint.h>

// LengthRegulator for MI455X (gfx1250): pure data-movement op (~320 MB HBM
// traffic, ~14us roofline at 23.3 TB/s). Strategy: stage each 4KB x-row into
// LDS with CDNA5 async global->LDS b128 loads (ASYNCcnt, NT hint: rows are
// read exactly once), gather from LDS, stream 256MB of output with 128-bit
// non-temporal stores. idx rows stay L2-resident (reused by 512 blocks each).

typedef float v4f __attribute__((ext_vector_type(4)));
typedef int   v4i __attribute__((ext_vector_type(4)));

#define B_    32
#define C_    512
#define T_    1024
#define MAXL  4096

// ---- CDNA5 async global->LDS (16B per lane), tracked by ASYNCcnt ----------
__device__ __forceinline__ void async_ld16(uint32_t lds_off, const void* g) {
  asm volatile("global_load_async_to_lds_b128 %0, %1, off"
               :: "v"(lds_off), "v"((unsigned long long)(uintptr_t)g)
               : "memory");
}
// Same, with non-temporal hint (read-once streaming data)
__device__ __forceinline__ void async_ld16_nt(uint32_t lds_off, const void* g) {
  asm volatile("global_load_async_to_lds_b128 %0, %1, off th:TH_LOAD_NT"
               :: "v"(lds_off), "v"((unsigned long long)(uintptr_t)g)
               : "memory");
}
__device__ __forceinline__ void wait_async0() {
  asm volatile("s_wait_asynccnt 0" ::: "memory");
}
// Low 32 bits of a generic pointer to __shared__ = LDS byte offset (aperture
// base lives in addr[63:32] on gfx1250).
__device__ __forceinline__ uint32_t lds_off(const void* p) {
  return (uint32_t)(uintptr_t)p;
}

// ---- Kernel 1: inclusive scan of durations; cum -> ws, mel_len -> out tail -
__global__ void lr_scan(const int* __restrict__ dur, int* __restrict__ cum,
                        float* __restrict__ mel_out) {
  __shared__ int s[T_];
  const int b = blockIdx.x;
  const int tid = threadIdx.x;
  s[tid] = dur[b * T_ + tid];
  __syncthreads();
  // Hillis-Steele inclusive scan (10 steps)
  for (int off = 1; off < T_; off <<= 1) {
    int add = (tid >= off) ? s[tid - off] : 0;
    __syncthreads();
    s[tid] += add;
    __syncthreads();
  }
  cum[b * T_ + tid] = s[tid];
  if (tid == T_ - 1) mel_out[b] = (float)s[tid];   // mel_len, as output dtype
}

// ---- Kernel 2: idx[b,t] = searchsorted(cum[b], t, 'right'); -1 if invalid --
// grid = (MAXL/256, B), block = 256
__global__ void lr_index(const int* __restrict__ cum, int* __restrict__ idxbuf) {
  __shared__ int cs[T_];
  const int b = blockIdx.y;
  const int tid = threadIdx.x;
  // stage the 4KB cum row: 256 lanes x 16B async b128 (reused -> default TH)
  async_ld16(lds_off(&cs[tid * 4]), cum + b * T_ + tid * 4);
  wait_async0();
  __syncthreads();

  const int t = blockIdx.x * 256 + tid;          // 0..4095
  const int mel = cs[T_ - 1];
  // upper-bound binary search over LDS
  int lo = 0, hi = T_;
  while (lo < hi) {
    int mid = (lo + hi) >> 1;
    if (cs[mid] <= t) lo = mid + 1; else hi = mid;
  }
  if (lo > T_ - 1) lo = T_ - 1;                  // clip (matches reference)
  idxbuf[b * MAXL + t] = (t < mel) ? lo : -1;    // fold validity into sign
}

// ---- Kernel 3: out[b,c,t] = idx>=0 ? x[b,c,idx] : 0 -----------------------
// grid = B*C blocks, block = 256; each block owns one x row (4KB) + 4096 t's
__global__ void lr_gather(const float* __restrict__ x,
                          const int* __restrict__ idxbuf,
                          float* __restrict__ out) {
  __shared__ float row[T_];
  const int bc  = blockIdx.x;           // b*C + c
  const int b   = bc >> 9;              // C_ == 512
  const int tid = threadIdx.x;

  // stage the 4KB x row once: 256 lanes x 16B async b128, non-temporal
  const float* src = x + (size_t)bc * T_;
  async_ld16_nt(lds_off(&row[tid * 4]), src + tid * 4);

  const v4i* __restrict__ ib = (const v4i*)(idxbuf + b * MAXL);
  v4f* __restrict__ ob = (v4f*)(out + (size_t)bc * MAXL);

  // Issue all idx loads now so their latency hides under the async staging.
  v4i i0 = ib[0 * 256 + tid];
  v4i i1 = ib[1 * 256 + tid];
  v4i i2 = ib[2 * 256 + tid];
  v4i i3 = ib[3 * 256 + tid];

  wait_async0();
  __syncthreads();

  v4f v;
  v.x = (i0.x >= 0) ? row[i0.x] : 0.0f;
  v.y = (i0.y >= 0) ? row[i0.y] : 0.0f;
  v.z = (i0.z >= 0) ? row[i0.z] : 0.0f;
  v.w = (i0.w >= 0) ? row[i0.w] : 0.0f;
  __builtin_nontemporal_store(v, ob + 0 * 256 + tid);

  v.x = (i1.x >= 0) ? row[i1.x] : 0.0f;
  v.y = (i1.y >= 0) ? row[i1.y] : 0.0f;
  v.z = (i1.z >= 0) ? row[i1.z] : 0.0f;
  v.w = (i1.w >= 0) ? row[i1.w] : 0.0f;
  __builtin_nontemporal_store(v, ob + 1 * 256 + tid);

  v.x = (i2.x >= 0) ? row[i2.x] : 0.0f;
  v.y = (i2.y >= 0) ? row[i2.y] : 0.0f;
  v.z = (i2.z >= 0) ? row[i2.z] : 0.0f;
  v.w = (i2.w >= 0) ? row[i2.w] : 0.0f;
  __builtin_nontemporal_store(v, ob + 2 * 256 + tid);

  v.x = (i3.x >= 0) ? row[i3.x] : 0.0f;
  v.y = (i3.y >= 0) ? row[i3.y] : 0.0f;
  v.z = (i3.z >= 0) ? row[i3.z] : 0.0f;
  v.w = (i3.w >= 0) ? row[i3.w] : 0.0f;
  __builtin_nontemporal_store(v, ob + 3 * 256 + tid);
}

extern "C" void kernel_launch(void* const* d_in, const int* in_sizes, int n_in,
                              void* d_out, int out_size, void* d_ws, size_t ws_size,
                              hipStream_t stream) {
  (void)in_sizes; (void)n_in; (void)out_size; (void)ws_size;

  const float* x   = (const float*)d_in[0];
  const int*   dur = (const int*)d_in[1];     // durations (harness int cast)
  // d_in[2] = max_len scalar (fixed 4096 per setup_inputs)

  float* out = (float*)d_out;
  float* mel_out = out + (size_t)B_ * C_ * MAXL;  // tail: 32 mel_len values

  int* cum    = (int*)d_ws;                   // 32*1024*4 = 128KB
  int* idxbuf = cum + B_ * T_;                // 32*4096*4 = 512KB

  lr_scan  <<<B_, T_, 0, stream>>>(dur, cum, mel_out);
  lr_index <<<dim3(MAXL / 256, B_), 256, 0, stream>>>(cum, idxbuf);
  lr_gather<<<B_ * C_, 256, 0, stream>>>(x, idxbuf, out);
}